// RecurrentDistribution_41832981463435
// MI455X (gfx1250) — compile-verified
//
#include <hip/hip_runtime.h>

typedef __attribute__((ext_vector_type(2))) float v2f;
typedef __attribute__((ext_vector_type(8))) float v8f;

#define B_   128
#define T_   1024
#define H_   256
#define HZ_  320
#define Z_   64
#define BT_  (B_ * T_)

// ---------------------------------------------------------------------------
// Phase 1 helpers: one k-step = A tile (16x4 f32) + 4 n-tiles of B for both
// weight matrices. A and B f32 tiles share the same per-lane float2 shape:
// lane<16 holds K+{0,1}, lane>=16 holds K+{2,3}.
// ---------------------------------------------------------------------------
__device__ __forceinline__ void load_kstep(int kt,
                                           const float* __restrict__ aPtr,
                                           const float* __restrict__ bMuPtr,
                                           const float* __restrict__ bStdPtr,
                                           v2f& a, v2f* bm, v2f* bs)
{
    const int k = kt * 4;
    // h is streamed exactly once -> non-temporal, keep L2 for pre-activations
    a = __builtin_nontemporal_load((const v2f*)(aPtr + k));
#pragma unroll
    for (int n = 0; n < 4; ++n) {
        // n-tile stride = 16 rows * 320 floats -> folds to immediate offsets
        bm[n] = *(const v2f*)(bMuPtr  + (size_t)n * 16 * HZ_ + k);
        bs[n] = *(const v2f*)(bStdPtr + (size_t)n * 16 * HZ_ + k);
    }
}

__device__ __forceinline__ void do_wmmas(v2f a, const v2f* bm, const v2f* bs,
                                         v8f* accMu, v8f* accStd)
{
#pragma unroll
    for (int n = 0; n < 4; ++n) {
        accMu[n]  = __builtin_amdgcn_wmma_f32_16x16x4_f32(
            false, a, false, bm[n], (short)0, accMu[n],  false, false);
        accStd[n] = __builtin_amdgcn_wmma_f32_16x16x4_f32(
            false, a, false, bs[n], (short)0, accStd[n], false, false);
    }
}

// ---------------------------------------------------------------------------
// Phase 1: hoisted (non-recurrent) part of both affine maps.
//   preMu [BT,64] = h[BT,256] @ Wmu[:, :256]^T + b_mu
//   preStd[BT,64] = h[BT,256] @ Wstd[:, :256]^T + b_std
// One wave per 16-row stripe; the wave owns all 4 N-tiles for BOTH matrices
// so h is read from HBM exactly once. Two-deep software pipeline keeps ~9
// loads in flight while 8 WMMAs consume the previous k-step.
// ---------------------------------------------------------------------------
__global__ __launch_bounds__(32) void RecurDist_phase1_gemm(
    const float* __restrict__ h,     // [BT, 256]
    const float* __restrict__ Wmu,   // [64, 320]
    const float* __restrict__ bmu,   // [64]
    const float* __restrict__ Wstd,  // [64, 320]
    const float* __restrict__ bstd,  // [64]
    float* __restrict__ preMu,       // [BT, 64]
    float* __restrict__ preStd)      // [BT, 64]
{
    const int lane    = threadIdx.x;      // 0..31
    const int lmod    = lane & 15;
    const int halfSel = lane >> 4;        // 0: K+{0,1}, 1: K+{2,3}
    const int rowBase = blockIdx.x * 16;

    v8f accMu[4], accStd[4];
#pragma unroll
    for (int n = 0; n < 4; ++n) { accMu[n] = (v8f){}; accStd[n] = (v8f){}; }

    const float* aPtr    = h    + (size_t)(rowBase + lmod) * H_ + 2 * halfSel;
    const float* bMuPtr  = Wmu  + (size_t)lmod * HZ_ + 2 * halfSel;
    const float* bStdPtr = Wstd + (size_t)lmod * HZ_ + 2 * halfSel;

    // two pipeline stages (distinct SSA values -> loads can batch per stage)
    v2f a0, bm0[4], bs0[4];
    v2f a1, bm1[4], bs1[4];

    load_kstep(0, aPtr, bMuPtr, bStdPtr, a0, bm0, bs0);
    load_kstep(1, aPtr, bMuPtr, bStdPtr, a1, bm1, bs1);

#pragma unroll 1
    for (int kt = 0; kt < (H_ / 4) - 2; kt += 2) {
        do_wmmas(a0, bm0, bs0, accMu, accStd);
        load_kstep(kt + 2, aPtr, bMuPtr, bStdPtr, a0, bm0, bs0);
        do_wmmas(a1, bm1, bs1, accMu, accStd);
        load_kstep(kt + 3, aPtr, bMuPtr, bStdPtr, a1, bm1, bs1);
    }
    do_wmmas(a0, bm0, bs0, accMu, accStd);
    do_wmmas(a1, bm1, bs1, accMu, accStd);

    // C/D layout: VGPR r -> M = r + 8*halfSel, lane%16 -> N
#pragma unroll
    for (int n = 0; n < 4; ++n) {
        const int col = n * 16 + lmod;
        const float bmv = bmu[col];
        const float bsv = bstd[col];
#pragma unroll
        for (int r = 0; r < 8; ++r) {
            const size_t row = (size_t)(rowBase + r + 8 * halfSel);
            preMu [row * Z_ + col] = accMu[n][r]  + bmv;
            preStd[row * Z_ + col] = accStd[n][r] + bsv;
        }
    }
}

// ---------------------------------------------------------------------------
// Phase 2: the serial recurrence. One workgroup (64 threads) per batch
// element; 128 independent scans run concurrently. Thread j owns output
// channel j: its two 64-entry recurrent weight rows live in VGPRs; z_{t-1}
// is double-buffered in LDS (conflict-free broadcast reads, one barrier per
// step). Reads pre-activations in place from the mu/std output slots and
// overwrites them with the final mu/std.
// ---------------------------------------------------------------------------
__global__ __launch_bounds__(64) void RecurDist_phase2_scan(
    const float* __restrict__ eps,    // [B, T, 64]
    const float* __restrict__ Wmu,    // [64, 320]
    const float* __restrict__ Wstd,   // [64, 320]
    float* __restrict__ outZ,         // [B, T, 64]
    float* __restrict__ outMu,        // in: preMu, out: mu
    float* __restrict__ outStd)       // in: preStd, out: std
{
    __shared__ float zbuf[2][Z_];
    const int j = threadIdx.x;        // 0..63 : output channel
    const int b = blockIdx.x;         // 0..127: batch element

    // Register-resident recurrent weights: W[:, 256:320] row j, both matrices.
    float wzm[Z_], wzs[Z_];
#pragma unroll
    for (int k = 0; k < Z_; ++k) {
        wzm[k] = Wmu [(size_t)j * HZ_ + H_ + k];
        wzs[k] = Wstd[(size_t)j * HZ_ + H_ + k];
    }

    zbuf[0][j] = 0.0f;                // z_0 = 0
    __syncthreads();

    const size_t base = (size_t)b * T_ * Z_ + j;
    int p = 0;
    for (int t = 0; t < T_; ++t) {
        const size_t idx = base + (size_t)t * Z_;
        // issue global loads early; consumed only after the dot-product loop
        const float amu  = outMu[idx];
        const float astd = outStd[idx];
        const float e    = eps[idx];

        float accm = 0.0f, accs = 0.0f;
#pragma unroll
        for (int k = 0; k < Z_; ++k) {
            const float zv = zbuf[p][k];          // LDS broadcast
            accm = fmaf(zv, wzm[k], accm);
            accs = fmaf(zv, wzs[k], accs);
        }

        const float mu = accm + amu;
        const float y  = accs + astd;
        // numerically stable softplus via v_exp_f32 / v_log_f32
        const float sp = fmaxf(y, 0.0f) + __logf(1.0f + __expf(-fabsf(y)));
        const float sd = sp + 1e-4f;
        const float zn = fmaf(e, sd, mu);

        zbuf[p ^ 1][j] = zn;                      // publish z_t for next step
        __builtin_nontemporal_store(zn, &outZ [idx]);
        __builtin_nontemporal_store(mu, &outMu[idx]);
        __builtin_nontemporal_store(sd, &outStd[idx]);

        __syncthreads();                          // one barrier per step
        p ^= 1;
    }
}

extern "C" void kernel_launch(void* const* d_in, const int* in_sizes, int n_in,
                              void* d_out, int out_size, void* d_ws, size_t ws_size,
                              hipStream_t stream) {
    (void)in_sizes; (void)n_in; (void)out_size; (void)d_ws; (void)ws_size;
    const float* h    = (const float*)d_in[0];  // input_q [128,1024,256]
    const float* eps  = (const float*)d_in[1];  // eps     [128,1024,64]
    const float* Wmu  = (const float*)d_in[2];  // [64,320]
    const float* bmu  = (const float*)d_in[3];  // [64]
    const float* Wstd = (const float*)d_in[4];  // [64,320]
    const float* bstd = (const float*)d_in[5];  // [64]

    float* outZ   = (float*)d_out;                  // [B,T,64]
    float* outMu  = outZ  + (size_t)BT_ * Z_;       // [B,T,64]
    float* outStd = outMu + (size_t)BT_ * Z_;       // [B,T,64]

    // Phase 1: parallel exact-f32 WMMA GEMM for the non-recurrent part.
    RecurDist_phase1_gemm<<<dim3(BT_ / 16), dim3(32), 0, stream>>>(
        h, Wmu, bmu, Wstd, bstd, outMu, outStd);

    // Phase 2: 128 concurrent latency-optimized scans (stream-ordered after
    // phase 1 on the same stream).
    RecurDist_phase2_scan<<<dim3(B_), dim3(64), 0, stream>>>(
        eps, Wmu, Wstd, outZ, outMu, outStd);
}